// EncoderRecurrent_10084583211237
// MI455X (gfx1250) — compile-verified
//
#include <hip/hip_runtime.h>
#include <math.h>

typedef __attribute__((ext_vector_type(2))) float v2f;
typedef __attribute__((ext_vector_type(4))) float v4f;
typedef __attribute__((ext_vector_type(8))) float v8f;

// ---------------------------------------------------------------------------
__global__ void mask_i2f(const int* __restrict__ mi, float* __restrict__ mf, int n) {
    int i = blockIdx.x * blockDim.x + threadIdx.x;
    if (i < n) mf[i] = (float)mi[i];
}

// stride-2 2x2x2 max-pool of occupancy mask
__global__ void down_mask(const float* __restrict__ mi, float* __restrict__ mo,
                          int B_, int Do) {
    int i = blockIdx.x * blockDim.x + threadIdx.x;
    int n = B_ * Do * Do * Do;
    if (i >= n) return;
    int x = i % Do; int t = i / Do;
    int y = t % Do; t /= Do;
    int z = t % Do; int b = t / Do;
    int Di = Do * 2;
    const float* base = mi + (long)b * Di * Di * Di;
    float m = 0.f;
    for (int dz = 0; dz < 2; ++dz)
        for (int dy = 0; dy < 2; ++dy)
            for (int dx = 0; dx < 2; ++dx)
                m = fmaxf(m, base[((long)(2 * z + dz) * Di + (2 * y + dy)) * Di + (2 * x + dx)]);
    mo[i] = m;
}

// NCDHW -> channels-last with fused mask:  o[b,vox,c] = x[b,c,vox] * m[b,vox]
__global__ void tx_mask(const float* __restrict__ x, const float* __restrict__ m,
                        float* __restrict__ o, int B_, int C, long S) {
    long i = (long)blockIdx.x * blockDim.x + threadIdx.x;
    long n = (long)B_ * S * C;
    if (i >= n) return;
    int c = (int)(i % C); long t = i / C;
    long vox = t % S; int b = (int)(t / S);
    o[i] = x[((long)b * C + c) * S + vox] * m[(long)b * S + vox];
}

// [Cout,Cin,k3] -> [Cout, k3*Cin] (K-major: K = sp*Cin + ci)
__global__ void pack_w(const float* __restrict__ w, float* __restrict__ wp,
                       int Cout, int Cin, int k3) {
    int i = blockIdx.x * blockDim.x + threadIdx.x;
    int n = Cout * Cin * k3;
    if (i >= n) return;
    int sp = i % k3; int t = i / k3;
    int ci = t % Cin; int co = t / Cin;
    wp[(long)co * (k3 * Cin) + sp * Cin + ci] = w[i];
}

// ---------------------------------------------------------------------------
// Implicit-GEMM 3D conv via V_WMMA_F32_16X16X4_F32.
// Wave tile: 32(co) x 16(voxels along x); two stacked accumulators share one
// B operand per K-step. K ordered (kz,ky,kx) outer / ci inner.
// inCL=1: activations channels-last [B,Di^3,Cin]; A and B operands are single
// aligned b64 loads at immediate offsets (no pointer math in the hot loop).
// inCL=0: NCDHW strided B path (used only for the tiny conv_h).
// Boundary taps: clamped in-bounds address + multiply by scale in {0,1}.
// modes: 0 = NCDHW store (acc+b)*m            (hx -> d_out)
//        1 = CL store relu(acc+b)*m
//        2 = CL store += (acc+b)*m            (residual)
//        3 = CL store sigmoid(relu(acc+b)*m)*aux*m
//        4 = CL store (acc+b)*m
__global__ __launch_bounds__(256)
void conv3d_wmma(const float* __restrict__ in,
                 const float* __restrict__ wt,      // packed [Cout, k3*Cin]
                 const float* __restrict__ bias,    // [Cout]
                 const float* __restrict__ maskOut, // [B, Do^3]
                 const float* __restrict__ aux,     // CL [B,Do^3,Cout] (mode 3)
                 float* __restrict__ out,
                 int B_, int Cin, int Cout, int Di, int Do,
                 int KS, int STRIDE, int PAD, int mode, int inCL)
{
    const int lane = threadIdx.x & 31;
    const int wave = blockIdx.x * (blockDim.x >> 5) + (threadIdx.x >> 5);
    const int xT = Do >> 4;
    const int nTiles = B_ * Do * Do * xT;
    const int coTiles = Cout >> 5;
    if (wave >= nTiles * coTiles) return;           // wave-uniform: EXEC all-ones

    const int coT = wave / nTiles;
    int nT = wave - coT * nTiles;
    const int x0 = (nT % xT) << 4;
    int t = nT / xT;
    const int y = t % Do; t /= Do;
    const int z = t % Do;
    const int b = t / Do;
    const int co0 = coT << 5;

    const int col  = lane & 15;
    const int koff = (lane >> 4) << 1;
    const int k3   = KS * KS * KS;
    const int Ktot = k3 * Cin;
    const long diS = (long)Di * Di * Di;
    const int DiSq = Di * Di;
    const long wRow0 = (long)(co0 + col) * Ktot;
    const long wRow1 = (long)(co0 + 16 + col) * Ktot;

    v8f acc0 = {};
    v8f acc1 = {};
    for (int kz = 0; kz < KS; ++kz) {
        const int iz = z * STRIDE + kz - PAD;
        const bool zv = (iz >= 0) & (iz < Di);
        const int ciz = min(max(iz, 0), Di - 1);
        for (int ky = 0; ky < KS; ++ky) {
            const int iy = y * STRIDE + ky - PAD;
            const bool yv = (iy >= 0) & (iy < Di);
            const int ciy = min(max(iy, 0), Di - 1);
            for (int kx = 0; kx < KS; ++kx) {
                const int sp = (kz * KS + ky) * KS + kx;
                const int ix = (x0 + col) * STRIDE + kx - PAD;
                const bool valid = zv & yv & (ix >= 0) & (ix < Di);
                const int cix = min(max(ix, 0), Di - 1);
                const long vox = (long)ciz * DiSq + (long)ciy * Di + cix;
                const float scale = valid ? 1.f : 0.f;
                const float* w0 = wt + wRow0 + (long)sp * Cin + koff;
                const float* w1 = wt + wRow1 + (long)sp * Cin + koff;
                if (inCL) {
                    const float* ip = in + ((long)b * diS + vox) * Cin + koff;
                    #pragma unroll 8
                    for (int ci0 = 0; ci0 < Cin; ci0 += 4) {
                        v2f a0 = *(const v2f*)(w0 + ci0);   // b64, imm offset
                        v2f a1 = *(const v2f*)(w1 + ci0);
                        v2f bv = *(const v2f*)(ip + ci0);
                        v2f bb; bb.x = bv.x * scale; bb.y = bv.y * scale;
                        acc0 = __builtin_amdgcn_wmma_f32_16x16x4_f32(
                            false, a0, false, bb, (short)0, acc0, false, false);
                        acc1 = __builtin_amdgcn_wmma_f32_16x16x4_f32(
                            false, a1, false, bb, (short)0, acc1, false, false);
                    }
                } else {
                    const float* ip = in + ((long)b * Cin + koff) * diS + vox;
                    #pragma unroll 4
                    for (int ci0 = 0; ci0 < Cin; ci0 += 4) {
                        v2f a0 = *(const v2f*)(w0 + ci0);
                        v2f a1 = *(const v2f*)(w1 + ci0);
                        v2f bb;
                        bb.x = ip[0]   * scale;
                        bb.y = ip[diS] * scale;
                        acc0 = __builtin_amdgcn_wmma_f32_16x16x4_f32(
                            false, a0, false, bb, (short)0, acc0, false, false);
                        acc1 = __builtin_amdgcn_wmma_f32_16x16x4_f32(
                            false, a1, false, bb, (short)0, acc1, false, false);
                        ip += 4 * diS;
                    }
                }
            }
        }
    }

    // Epilogue. C layout: lane l, VGPR v -> (M = v + (l>>4)*8, N = l&15)
    const long doS = (long)Do * Do * Do;
    const long voxOut = (long)z * Do * Do + (long)y * Do + (x0 + col);
    const float mo = maskOut[(long)b * doS + voxOut];
    const int rbase = (lane >> 4) << 3;
    const long cbase = ((long)b * doS + voxOut) * Cout;
    for (int half = 0; half < 2; ++half) {
        const v8f acc = half ? acc1 : acc0;
        const int coB = co0 + half * 16 + rbase;      // 8 consecutive channels
        if (mode == 0) {                               // NCDHW strided (hx)
            for (int v = 0; v < 8; ++v)
                out[((long)b * Cout + coB + v) * doS + voxOut] =
                    (acc[v] + bias[coB + v]) * mo;
            continue;
        }
        v4f blo = *(const v4f*)(bias + coB);
        v4f bhi = *(const v4f*)(bias + coB + 4);
        float vals[8];
        for (int v = 0; v < 8; ++v) {
            float val = acc[v] + (v < 4 ? blo[v] : bhi[v - 4]);
            if (mode == 1)      val = fmaxf(val, 0.f) * mo;
            else if (mode == 4) val = val * mo;
            else if (mode == 2) val = val * mo;        // added below
            else {                                     // mode 3: gate
                float g = fmaxf(val, 0.f) * mo;
                val = (1.f / (1.f + __expf(-g))) * mo; // * aux below
            }
            vals[v] = val;
        }
        float* op = out + cbase + coB;
        v4f lo = {vals[0], vals[1], vals[2], vals[3]};
        v4f hi = {vals[4], vals[5], vals[6], vals[7]};
        if (mode == 2) {
            v4f o0 = *(const v4f*)op;
            v4f o1 = *(const v4f*)(op + 4);
            lo += o0; hi += o1;
        } else if (mode == 3) {
            const float* ap = aux + cbase + coB;
            v4f a0 = *(const v4f*)ap;
            v4f a1 = *(const v4f*)(ap + 4);
            lo *= a0; hi *= a1;
        }
        *(v4f*)op = lo;
        *(v4f*)(op + 4) = hi;
    }
}

// ---------------------------------------------------------------------------
static inline void launch_conv(hipStream_t s,
                               const float* in, const float* wt, const float* bi,
                               const float* mOut, const float* aux,
                               float* out, int B_, int Cin, int Cout,
                               int Di, int Do, int KS, int ST, int PAD,
                               int mode, int inCL) {
    int waves = (Cout >> 5) * (B_ * Do * Do * (Do >> 4));
    int blocks = (waves + 7) / 8;
    conv3d_wmma<<<blocks, 256, 0, s>>>(in, wt, bi, mOut, aux, out,
                                       B_, Cin, Cout, Di, Do, KS, ST, PAD, mode, inCL);
}

extern "C" void kernel_launch(void* const* d_in, const int* in_sizes, int n_in,
                              void* d_out, int out_size, void* d_ws, size_t ws_size,
                              hipStream_t stream) {
    const float* x     = (const float*)d_in[0];
    const int*   maski = (const int*)d_in[1];
    // d_in[2] = coder_num (== 2 in reference)
    const float* w_out = (const float*)d_in[3];
    const float* b_out = (const float*)d_in[4];
    const float* w_h   = (const float*)d_in[5];
    const float* b_h   = (const float*)d_in[6];
    const float* w_c0  = (const float*)d_in[7];
    const float* b_c0  = (const float*)d_in[8];
    const float* w_r0a = (const float*)d_in[9];
    const float* b_r0a = (const float*)d_in[10];
    const float* w_r0b = (const float*)d_in[11];
    const float* b_r0b = (const float*)d_in[12];
    const float* w_c1  = (const float*)d_in[13];
    const float* b_c1  = (const float*)d_in[14];

    const int B_ = 2;
    // workspace layout (floats)
    float* ws     = (float*)d_ws;
    float* m0     = ws;                    // 2*64^3  = 524288
    float* m1     = m0 + 524288;           // 2*32^3  = 65536
    float* m2     = m1 + 65536;            // 2*16^3  = 8192
    float* pw_out = m2 + 8192;             // 32*1728 = 55296
    float* pw_h   = pw_out + 55296;        // 64*256  = 16384
    float* pw_c0  = pw_h + 16384;          // 64*512  = 32768
    float* pw_r0a = pw_c0 + 32768;         // 64*1728 = 110592
    float* pw_r0b = pw_r0a + 110592;       // 110592
    float* pw_c1  = pw_r0b + 110592;       // 110592
    float* bufX   = pw_c1 + 110592;        // CL masked x: 2*64^3*64 = 33554432
    float* bufC   = bufX + 33554432;       // CL 2*32^3*64 = 4194304
    float* bufR   = bufC + 4194304;
    float* bufCx  = bufR + 4194304;

    float* out0 = (float*)d_out;           // NCDHW [2,32,64^3]
    float* out1 = out0 + (long)2 * 32 * 64 * 64 * 64;
    float* out2 = out1 + (long)2 * 32 * 32 * 32 * 32;

    // masks
    { int n = 2 * 64 * 64 * 64; mask_i2f<<<(n + 255) / 256, 256, 0, stream>>>(maski, m0, n); }
    { int n = 2 * 32 * 32 * 32; down_mask<<<(n + 255) / 256, 256, 0, stream>>>(m0, m1, B_, 32); }
    { int n = 2 * 16 * 16 * 16; down_mask<<<(n + 255) / 256, 256, 0, stream>>>(m1, m2, B_, 16); }
    // weight repack (K-major)
    { int n = 32*64*27; pack_w<<<(n+255)/256, 256, 0, stream>>>(w_out, pw_out, 32, 64, 27); }
    { int n = 64*32*8;  pack_w<<<(n+255)/256, 256, 0, stream>>>(w_h,   pw_h,   64, 32, 8);  }
    { int n = 64*64*8;  pack_w<<<(n+255)/256, 256, 0, stream>>>(w_c0,  pw_c0,  64, 64, 8);  }
    { int n = 64*64*27; pack_w<<<(n+255)/256, 256, 0, stream>>>(w_r0a, pw_r0a, 64, 64, 27); }
    { int n = 64*64*27; pack_w<<<(n+255)/256, 256, 0, stream>>>(w_r0b, pw_r0b, 64, 64, 27); }
    { int n = 64*64*27; pack_w<<<(n+255)/256, 256, 0, stream>>>(w_c1,  pw_c1,  64, 64, 27); }
    // cx0 = x*m0, channels-last
    { long n = (long)2 * 262144 * 64;
      tx_mask<<<(int)((n + 255) / 256), 256, 0, stream>>>(x, m0, bufX, B_, 64, 262144); }

    // hx0 = conv_out(cx0)*m0 -> d_out (NCDHW)
    launch_conv(stream, bufX, pw_out, b_out, m0, nullptr, out0,
                B_, 64, 32, 64, 64, 3, 1, 1, 0, 1);

    // ---------------- iteration 1: 64 -> 32 ----------------
    launch_conv(stream, bufX,  pw_c0,  b_c0,  m1, nullptr, bufC,
                B_, 64, 64, 64, 32, 2, 2, 0, 1, 1);   // c = relu(conv)*m
    launch_conv(stream, bufC,  pw_r0a, b_r0a, m1, nullptr, bufR,
                B_, 64, 64, 32, 32, 3, 1, 1, 1, 1);   // r = relu(conv)*m
    launch_conv(stream, bufR,  pw_r0b, b_r0b, m1, nullptr, bufC,
                B_, 64, 64, 32, 32, 3, 1, 1, 2, 1);   // c += conv(r)*m
    launch_conv(stream, bufC,  pw_c1,  b_c1,  m1, nullptr, bufR,
                B_, 64, 64, 32, 32, 3, 1, 1, 4, 1);   // c = conv(c)*m
    launch_conv(stream, out0,  pw_h,   b_h,   m1, bufR, bufCx,
                B_, 32, 64, 64, 32, 2, 2, 0, 3, 0);   // cx = sigm(relu(conv(hx))*m)*c*m
    launch_conv(stream, bufCx, pw_out, b_out, m1, nullptr, out1,
                B_, 64, 32, 32, 32, 3, 1, 1, 0, 1);   // hx1 -> d_out

    // ---------------- iteration 2: 32 -> 16 ----------------
    launch_conv(stream, bufCx, pw_c0,  b_c0,  m2, nullptr, bufC,
                B_, 64, 64, 32, 16, 2, 2, 0, 1, 1);
    launch_conv(stream, bufC,  pw_r0a, b_r0a, m2, nullptr, bufR,
                B_, 64, 64, 16, 16, 3, 1, 1, 1, 1);
    launch_conv(stream, bufR,  pw_r0b, b_r0b, m2, nullptr, bufC,
                B_, 64, 64, 16, 16, 3, 1, 1, 2, 1);
    launch_conv(stream, bufC,  pw_c1,  b_c1,  m2, nullptr, bufR,
                B_, 64, 64, 16, 16, 3, 1, 1, 4, 1);
    launch_conv(stream, out1,  pw_h,   b_h,   m2, bufR, bufCx,
                B_, 32, 64, 32, 16, 2, 2, 0, 3, 0);
    launch_conv(stream, bufCx, pw_out, b_out, m2, nullptr, out2,
                B_, 64, 32, 16, 16, 3, 1, 1, 0, 1);
}